// RelationalGraphEncoder_35021163331782
// MI455X (gfx1250) — compile-verified
//
#include <hip/hip_runtime.h>
#include <hip/hip_bf16.h>
#include <stdint.h>

// ---------------- problem constants (match reference) ----------------
#define Nn 50000
#define Ee 1000000
#define Rr 16
#define NBb 16
#define Dd 128
#define Ll 3
#define Gg 64
#define LN_EPS 1e-5f

typedef __attribute__((ext_vector_type(2))) float v2f;
typedef __attribute__((ext_vector_type(8))) float v8f;

// ---------------- small utility kernels ----------------
__global__ void k_zero_f32(float* __restrict__ p, int n4) {
    int i = blockIdx.x * blockDim.x + threadIdx.x;
    if (i < n4) { float4 z = {0.f, 0.f, 0.f, 0.f}; ((float4*)p)[i] = z; }
}

__global__ void k_zero_i32(int* __restrict__ p, int n) {
    int i = blockIdx.x * blockDim.x + threadIdx.x;
    if (i < n) p[i] = 0;
}

// counts per (dst,relation) segment and per-relation histogram
__global__ void k_count(const int* __restrict__ ei, const int* __restrict__ et,
                        int* __restrict__ cnt_nr, int* __restrict__ rel_hist) {
    int e = blockIdx.x * blockDim.x + threadIdx.x;
    if (e >= Ee) return;
    int d = ei[Ee + e];
    int t = et[e];
    atomicAdd(&cnt_nr[d * Rr + t], 1);
    atomicAdd(&rel_hist[t], 1);
}

__global__ void k_invcnt(const int* __restrict__ cnt_nr, float* __restrict__ inv) {
    int i = blockIdx.x * blockDim.x + threadIdx.x;
    if (i >= Nn * Rr) return;
    inv[i] = 1.0f / fmaxf((float)cnt_nr[i], 1.0f);
}

// serial 16-element exclusive prefix over relation histogram
__global__ void k_prefix(const int* __restrict__ rel_hist, int* __restrict__ rel_off,
                         int* __restrict__ cursor) {
    if (threadIdx.x == 0 && blockIdx.x == 0) {
        int acc = 0;
        for (int r = 0; r < Rr; ++r) { rel_off[r] = acc; cursor[r] = acc; acc += rel_hist[r]; }
        rel_off[Rr] = acc;
    }
}

// bucket-sort edge ids by relation
__global__ void k_fill_sorted(const int* __restrict__ et, int* __restrict__ cursor,
                              int* __restrict__ sorted) {
    int e = blockIdx.x * blockDim.x + threadIdx.x;
    if (e >= Ee) return;
    int t = et[e];
    int p = atomicAdd(&cursor[t], 1);
    sorted[p] = e;
}

// x = emb[node_ids]   (one wave per node, float4 per lane)
__global__ void k_gather(const int* __restrict__ node_ids, const float* __restrict__ emb,
                         float* __restrict__ x) {
    int g = blockIdx.x * blockDim.x + threadIdx.x;
    int row = g >> 5, lane = g & 31;
    if (row >= Nn) return;
    int v = node_ids[row];
    ((float4*)(x + (size_t)row * Dd))[lane] = ((const float4*)(emb + (size_t)v * Dd))[lane];
}

// W[r,i,o] = sum_b comp[r,b] * basis[b,i,o]
__global__ void k_build_w(const float* __restrict__ comp_l, const float* __restrict__ basis_l,
                          float* __restrict__ W) {
    int idx = blockIdx.x * blockDim.x + threadIdx.x;      // R*D*D = 262144
    if (idx >= Rr * Dd * Dd) return;
    int r = idx >> 14;            // /(D*D)
    int io = idx & (Dd * Dd - 1);
    float acc = 0.f;
#pragma unroll
    for (int b = 0; b < NBb; ++b) acc += comp_l[r * NBb + b] * basis_l[b * Dd * Dd + io];
    W[idx] = acc;
}

// per-relation scatter: agg[dst] += x[src] for edges of relation r (pre-bucketed)
__global__ void k_scatter_rel(const int* __restrict__ sorted, const int* __restrict__ rel_off,
                              int r, const int* __restrict__ ei, const float* __restrict__ x,
                              float* __restrict__ agg) {
    int g = blockIdx.x * blockDim.x + threadIdx.x;
    int w = g >> 5, lane = g & 31;
    int lo = rel_off[r], hi = rel_off[r + 1];
    int ei_idx = lo + w;
    if (ei_idx >= hi) return;
    int e = sorted[ei_idx];
    int s = ei[e];          // src
    int d = ei[Ee + e];     // dst
    float4 v = ((const float4*)(x + (size_t)s * Dd))[lane];
    float* dstp = agg + (size_t)d * Dd + lane * 4;
    atomicAdd(dstp + 0, v.x);
    atomicAdd(dstp + 1, v.y);
    atomicAdd(dstp + 2, v.z);
    atomicAdd(dstp + 3, v.w);
}

// ---------------- WMMA GEMM: OUT[M,128] (+)= (rowscale .* A[M,128]) @ B[128,128] (+ bias) ----
// Block: 256 threads = 8 waves; block covers 128 rows; wave covers 16 rows x all 128 cols.
// V_WMMA_F32_16X16X4_F32 (fp32 in / fp32 acc) -- exact-precision path.
// B tile staged to LDS with GLOBAL_LOAD_ASYNC_TO_LDS_B128 (ASYNCcnt), A rows prefetched
// concurrently with global_prefetch_b8.
__global__ __launch_bounds__(256)
void k_gemm128_wmma(const float* __restrict__ A, const float* __restrict__ B,
                    float* __restrict__ OUT, const float* __restrict__ rowscale,
                    int scale_stride, const float* __restrict__ bias, int M, int accumulate) {
    __shared__ float Bs[Dd * Dd];                   // 64 KB: B tile in LDS
    const int tid = threadIdx.x;
    const int wave = tid >> 5;
    const int lane = tid & 31;
    const int m0 = blockIdx.x * 128 + wave * 16;
    const int lhalf = lane >> 4;                    // 0: lanes 0-15, 1: lanes 16-31
    const int l16 = lane & 15;
    const int arow = m0 + l16;                      // A row handled by this lane
    const float* Arow = A + (size_t)arow * Dd;

    // --- async stage B (global -> LDS, no VGPR round trip), 256 thr x 16 x 16B = 64 KB ---
    {
        const uint32_t lds_base = (uint32_t)(uintptr_t)&Bs[0];  // flat->LDS: low 32 bits
        for (int i = tid * 4; i < Dd * Dd; i += 256 * 4) {
            uint32_t loff = lds_base + (uint32_t)i * 4u;
            uint64_t ga = (uint64_t)(uintptr_t)(B + i);
            asm volatile("global_load_async_to_lds_b128 %0, %1, off"
                         :: "v"(loff), "v"(ga) : "memory");
        }
    }

    // --- overlap: prefetch this wave's 16 A rows (2 x 256B per lane covers 512B row) ---
    if (m0 < M) {
        __builtin_prefetch(Arow + lhalf * 64, 0, 3);
        __builtin_prefetch(Arow + 32 + lhalf * 64, 0, 3);
    }

    asm volatile("s_wait_asynccnt 0" ::: "memory");
    __syncthreads();

    if (m0 >= M) return;                            // wave-uniform: EXEC all-ones for WMMA

    float s = 1.0f;
    if (rowscale) s = rowscale[(size_t)arow * scale_stride];

    v8f acc[8];
    if (accumulate) {
#pragma unroll
        for (int j = 0; j < 8; ++j)
#pragma unroll
            for (int v = 0; v < 8; ++v)
                acc[j][v] = OUT[(size_t)(m0 + v + 8 * lhalf) * Dd + j * 16 + l16];
    } else if (bias) {
#pragma unroll
        for (int j = 0; j < 8; ++j) {
            float bv = bias[j * 16 + l16];
#pragma unroll
            for (int v = 0; v < 8; ++v) acc[j][v] = bv;
        }
    } else {
#pragma unroll
        for (int j = 0; j < 8; ++j)
#pragma unroll
            for (int v = 0; v < 8; ++v) acc[j][v] = 0.f;
    }

    for (int k0 = 0; k0 < Dd; k0 += 4) {
        const int ka = k0 + 2 * lhalf;
        v2f a;                                      // A frag: lane=M, VGPR0/1 = K pair
        a[0] = Arow[ka] * s;
        a[1] = Arow[ka + 1] * s;
#pragma unroll
        for (int j = 0; j < 8; ++j) {               // 8 column tiles
            v2f b;                                  // B frag: lane=N, VGPR0/1 = K pair
            b[0] = Bs[ka * Dd + j * 16 + l16];
            b[1] = Bs[(ka + 1) * Dd + j * 16 + l16];
            acc[j] = __builtin_amdgcn_wmma_f32_16x16x4_f32(
                false, a, false, b, (short)0, acc[j], false, false);
        }
    }

#pragma unroll
    for (int j = 0; j < 8; ++j)
#pragma unroll
        for (int v = 0; v < 8; ++v)
            OUT[(size_t)(m0 + v + 8 * lhalf) * Dd + j * 16 + l16] = acc[j][v];
}

// ---------------- fused LayerNorm + ReLU + residual: x += relu(LN(out)) ----------------
__global__ void k_ln_relu_res(const float* __restrict__ out, float* __restrict__ x,
                              const float* __restrict__ gamma_l, const float* __restrict__ beta_l) {
    int g = blockIdx.x * blockDim.x + threadIdx.x;
    int row = g >> 5, lane = g & 31;
    if (row >= Nn) return;
    float4 v = ((const float4*)(out + (size_t)row * Dd))[lane];
    float sum = v.x + v.y + v.z + v.w;
#pragma unroll
    for (int m = 16; m >= 1; m >>= 1) sum += __shfl_xor(sum, m, 32);
    float mu = sum * (1.0f / Dd);
    float dx = v.x - mu, dy = v.y - mu, dz = v.z - mu, dw = v.w - mu;
    float ss = dx * dx + dy * dy + dz * dz + dw * dw;
#pragma unroll
    for (int m = 16; m >= 1; m >>= 1) ss += __shfl_xor(ss, m, 32);
    float inv = rsqrtf(ss * (1.0f / Dd) + LN_EPS);
    int c = lane * 4;
    float4 gm = *(const float4*)(gamma_l + c);
    float4 bt = *(const float4*)(beta_l + c);
    float4 xo = ((const float4*)(x + (size_t)row * Dd))[lane];
    xo.x += fmaxf(dx * inv * gm.x + bt.x, 0.f);
    xo.y += fmaxf(dy * inv * gm.y + bt.y, 0.f);
    xo.z += fmaxf(dz * inv * gm.z + bt.z, 0.f);
    xo.w += fmaxf(dw * inv * gm.w + bt.w, 0.f);
    ((float4*)(x + (size_t)row * Dd))[lane] = xo;
}

// ---------------- pooling & output ----------------
__global__ void k_copy_x(const float* __restrict__ x, float* __restrict__ out) {
    int i = blockIdx.x * blockDim.x + threadIdx.x;
    if (i < Nn * Dd / 4) ((float4*)out)[i] = ((const float4*)x)[i];
}

__global__ void k_pool_sum(const float* __restrict__ x, const int* __restrict__ batch,
                           float* __restrict__ gsum, float* __restrict__ gcnt) {
    int g = blockIdx.x * blockDim.x + threadIdx.x;
    int row = g >> 5, lane = g & 31;
    if (row >= Nn) return;
    int b = batch[row];
    float4 v = ((const float4*)(x + (size_t)row * Dd))[lane];
    float* dp = gsum + (size_t)b * Dd + lane * 4;
    atomicAdd(dp + 0, v.x);
    atomicAdd(dp + 1, v.y);
    atomicAdd(dp + 2, v.z);
    atomicAdd(dp + 3, v.w);
    if (lane == 0) atomicAdd(&gcnt[b], 1.0f);
}

__global__ void k_pool_div(const float* __restrict__ gsum, const float* __restrict__ gcnt,
                           float* __restrict__ ge) {
    int i = blockIdx.x * blockDim.x + threadIdx.x;
    if (i >= Gg * Dd) return;
    ge[i] = gsum[i] / fmaxf(gcnt[i >> 7], 1.0f);
}

// ---------------- host orchestration ----------------
extern "C" void kernel_launch(void* const* d_in, const int* in_sizes, int n_in,
                              void* d_out, int out_size, void* d_ws, size_t ws_size,
                              hipStream_t stream) {
    const int*   node_ids = (const int*)d_in[0];
    const int*   ei       = (const int*)d_in[1];   // [2,E]: src then dst
    const int*   et       = (const int*)d_in[2];
    const int*   batch    = (const int*)d_in[3];
    const float* emb      = (const float*)d_in[4]; // [V,D]
    const float* comp     = (const float*)d_in[5]; // [L,R,NB]
    const float* basis    = (const float*)d_in[6]; // [L,NB,D,D]
    const float* root     = (const float*)d_in[7]; // [L,D,D]
    const float* bias     = (const float*)d_in[8]; // [L,D]
    const float* gamma    = (const float*)d_in[9]; // [L,D]
    const float* beta     = (const float*)d_in[10];// [L,D]

    float* outx = (float*)d_out;            // [N,D]
    float* outg = outx + (size_t)Nn * Dd;   // [G,D]

    // workspace layout (floats, then ints)
    float* f = (float*)d_ws;
    float* x    = f;                        f += (size_t)Nn * Dd;    // 6.4M
    float* out  = f;                        f += (size_t)Nn * Dd;    // 6.4M
    float* agg  = f;                        f += (size_t)Nn * Dd;    // 6.4M
    float* W    = f;                        f += (size_t)Rr * Dd * Dd; // 262144
    float* inv  = f;                        f += (size_t)Nn * Rr;    // 800000
    float* gsum = f;                        f += (size_t)Gg * Dd;    // 8192
    float* gcnt = f;                        f += Gg;                 // 64
    int* ip = (int*)f;
    int* cnt_nr   = ip;                     ip += Nn * Rr;           // 800000
    int* rel_hist = ip;                     ip += Rr;
    int* rel_off  = ip;                     ip += Rr + 1;
    int* cursor   = ip;                     ip += Rr;
    int* sorted   = ip;                     ip += Ee;                // 1M

    const int T = 256;
    // ---- edge preprocessing (counts, inv_cnt, bucket sort by relation) ----
    k_zero_i32<<<(Nn * Rr + T - 1) / T, T, 0, stream>>>(cnt_nr, Nn * Rr);
    k_zero_i32<<<1, T, 0, stream>>>(rel_hist, Rr);
    k_count<<<(Ee + T - 1) / T, T, 0, stream>>>(ei, et, cnt_nr, rel_hist);
    k_invcnt<<<(Nn * Rr + T - 1) / T, T, 0, stream>>>(cnt_nr, inv);
    k_prefix<<<1, 1, 0, stream>>>(rel_hist, rel_off, cursor);
    k_fill_sorted<<<(Ee + T - 1) / T, T, 0, stream>>>(et, cursor, sorted);

    // ---- embedding gather ----
    k_gather<<<(Nn * 32 + T - 1) / T, T, 0, stream>>>(node_ids, emb, x);

    const int gemm_blocks = (Nn + 127) / 128;              // 391
    const int scatter_blocks = (Ee * 32 + T - 1) / T;      // 125000 (covers worst-case bucket)
    const int rowk_blocks = (Nn * 32 + T - 1) / T;         // 6250

    for (int l = 0; l < Ll; ++l) {
        // W_r = comp[l] x basis[l]
        k_build_w<<<(Rr * Dd * Dd + T - 1) / T, T, 0, stream>>>(
            comp + (size_t)l * Rr * NBb, basis + (size_t)l * NBb * Dd * Dd, W);
        // out = x @ root_l + bias_l
        k_gemm128_wmma<<<gemm_blocks, T, 0, stream>>>(
            x, root + (size_t)l * Dd * Dd, out, nullptr, 0, bias + (size_t)l * Dd, Nn, 0);
        // out += mean_r @ W_r  for each relation
        for (int r = 0; r < Rr; ++r) {
            k_zero_f32<<<(Nn * Dd / 4 + T - 1) / T, T, 0, stream>>>(agg, Nn * Dd / 4);
            k_scatter_rel<<<scatter_blocks, T, 0, stream>>>(sorted, rel_off, r, ei, x, agg);
            k_gemm128_wmma<<<gemm_blocks, T, 0, stream>>>(
                agg, W + (size_t)r * Dd * Dd, out, inv + r, Rr, nullptr, Nn, 1);
        }
        // x += relu(LN(out))
        k_ln_relu_res<<<rowk_blocks, T, 0, stream>>>(
            out, x, gamma + (size_t)l * Dd, beta + (size_t)l * Dd);
    }

    // ---- outputs: x then graph mean-pool ----
    k_copy_x<<<(Nn * Dd / 4 + T - 1) / T, T, 0, stream>>>(x, outx);
    k_zero_f32<<<(Gg * Dd / 4 + T - 1) / T, T, 0, stream>>>(gsum, Gg * Dd / 4);
    k_zero_f32<<<1, T, 0, stream>>>(gcnt, Gg / 4);
    k_pool_sum<<<rowk_blocks, T, 0, stream>>>(x, batch, gsum, gcnt);
    k_pool_div<<<(Gg * Dd + T - 1) / T, T, 0, stream>>>(gsum, gcnt, outg);
}